// HungarianMatcher_32375463477430
// MI455X (gfx1250) — compile-verified
//
#include <hip/hip_runtime.h>
#include <hip/hip_bf16.h>
#include <math.h>

#define Qn   600    // queries
#define QPAD 608    // padded to multiple of 16 for WMMA M tiles
#define Nn   80     // targets (5 x 16 N-tiles)
#define Cc   92     // classes (23 x K=4 WMMA steps)
#define NTl  256    // LSA block size (8 waves)

typedef float v2f __attribute__((ext_vector_type(2)));
typedef float v8f __attribute__((ext_vector_type(8)));

// ---------------------------------------------------------------------------
// Kernel 1: row softmax of pred_logits -> probs (B, QPAD, 92), pad rows = 0
// One wave per (b,q) row; 92 classes = 3 strided lane loads.
// ---------------------------------------------------------------------------
__global__ void hm_softmax_kernel(const float* __restrict__ logits,
                                  float* __restrict__ probs, int B)
{
    const int wavesPerBlock = blockDim.x >> 5;
    const int row  = blockIdx.x * wavesPerBlock + (threadIdx.x >> 5);
    const int lane = threadIdx.x & 31;
    const int total = B * QPAD;
    if (row >= total) return;                       // wave-uniform
    const int b = row / QPAD;
    const int q = row - b * QPAD;
    float* out = probs + (size_t)row * Cc;
    if (q >= Qn) {                                  // zero the WMMA pad rows
        for (int c = lane; c < Cc; c += 32) out[c] = 0.0f;
        return;
    }
    const float* in = logits + ((size_t)b * Qn + q) * Cc;
    float x0 = (lane      < Cc) ? in[lane]      : -3.4e38f;
    float x1 = (lane + 32 < Cc) ? in[lane + 32] : -3.4e38f;
    float x2 = (lane + 64 < Cc) ? in[lane + 64] : -3.4e38f;
    float m = fmaxf(x0, fmaxf(x1, x2));
    #pragma unroll
    for (int off = 16; off; off >>= 1) m = fmaxf(m, __shfl_xor(m, off, 32));
    float e0 = (lane      < Cc) ? expf(x0 - m) : 0.0f;
    float e1 = (lane + 32 < Cc) ? expf(x1 - m) : 0.0f;
    float e2 = (lane + 64 < Cc) ? expf(x2 - m) : 0.0f;
    float s = e0 + e1 + e2;
    #pragma unroll
    for (int off = 16; off; off >>= 1) s += __shfl_xor(s, off, 32);
    float inv = 1.0f / s;
    if (lane      < Cc) out[lane]      = e0 * inv;
    if (lane + 32 < Cc) out[lane + 32] = e1 * inv;
    if (lane + 64 < Cc) out[lane + 64] = e2 * inv;
}

// ---------------------------------------------------------------------------
// Kernel 2: cost matrix build, transposed (B, N, Q).
// cost_class via V_WMMA_F32_16X16X4_F32 against one-hot(label) B-fragments
// (exact: one nonzero product per output element), fused L1 + GIoU epilogue.
// One wave per (b, 16-q-tile, 16-n-tile): 8 accumulator VGPRs, no spills.
// ---------------------------------------------------------------------------
__global__ void hm_cost_kernel(const float* __restrict__ probs,
                               const float* __restrict__ pred_boxes,
                               const int*   __restrict__ tgt_labels,
                               const float* __restrict__ tgt_boxes,
                               float* __restrict__ costT, int B)
{
    const int wavesPerBlock = blockDim.x >> 5;
    const int wid  = blockIdx.x * wavesPerBlock + (threadIdx.x >> 5);
    const int lane = threadIdx.x & 31;
    const int numQT = QPAD / 16;                    // 38
    const int numNT = Nn / 16;                      // 5
    if (wid >= B * numQT * numNT) return;           // wave-uniform (EXEC stays full)
    const int b   = wid / (numQT * numNT);
    const int rem = wid - b * (numQT * numNT);
    const int qt  = rem / numNT;
    const int nt  = rem - qt * numNT;
    const int q0   = qt * 16;
    const int half = lane >> 4;                     // 0: K lo / M 0-7 ; 1: K hi / M 8-15
    const int l15  = lane & 15;
    const int n    = nt * 16 + l15;

    const int lab = tgt_labels[b * Nn + n];

    v8f acc = (v8f)0.0f;
    const float* Pb = probs + ((size_t)b * QPAD + q0) * Cc + (size_t)l15 * Cc;
    #pragma unroll
    for (int k = 0; k < Cc; k += 4) {
        const int kk = k + half * 2;
        // A (16x4 f32): lane&15 = M row; VGPR0/1 hold K = kk, kk+1
        v2f a = *reinterpret_cast<const v2f*>(Pb + kk);
        // B (4x16 f32) one-hot of target label: lane&15 = N col
        v2f bf;
        bf.x = (lab == kk    ) ? 1.0f : 0.0f;
        bf.y = (lab == kk + 1) ? 1.0f : 0.0f;
        acc = __builtin_amdgcn_wmma_f32_16x16x4_f32(
            false, a, false, bf, (short)0, acc, false, false);
    }

    // Epilogue: acc element r maps to (q = q0 + half*8 + r, n)
    const int q_base = q0 + half * 8;
    const float4 tb = reinterpret_cast<const float4*>(tgt_boxes)[(size_t)b * Nn + n];
    const float ta = (tb.z - tb.x) * (tb.w - tb.y);
    const float* ap = reinterpret_cast<const float*>(&acc);
    float* crow = costT + ((size_t)b * Nn + n) * Qn;
    #pragma unroll
    for (int r = 0; r < 8; ++r) {
        const int q = q_base + r;
        if (q >= Qn) continue;
        const float4 pb = reinterpret_cast<const float4*>(pred_boxes)[(size_t)b * Qn + q];
        const float pa = (pb.z - pb.x) * (pb.w - pb.y);
        const float l1 = fabsf(pb.x - tb.x) + fabsf(pb.y - tb.y)
                       + fabsf(pb.z - tb.z) + fabsf(pb.w - tb.w);
        float iw = fminf(pb.z, tb.z) - fmaxf(pb.x, tb.x); iw = fmaxf(iw, 0.0f);
        float ih = fminf(pb.w, tb.w) - fmaxf(pb.y, tb.y); ih = fmaxf(ih, 0.0f);
        const float inter = iw * ih;
        const float uni   = pa + ta - inter;
        const float iou   = inter / uni;
        float cw = fmaxf(pb.z, tb.z) - fminf(pb.x, tb.x); cw = fmaxf(cw, 0.0f);
        float ch = fmaxf(pb.w, tb.w) - fminf(pb.y, tb.y); ch = fmaxf(ch, 0.0f);
        const float ac   = cw * ch;
        const float giou = iou - (ac - uni) / ac;
        crow[q] = -ap[r] + 5.0f * l1 - 2.0f * giou;   // 1*cls + 5*l1 + 2*(-giou)
    }
}

// ---------------------------------------------------------------------------
// Kernel 3: block-parallel Jonker-Volgenant LSA on costT[b] (80 x 600),
// one 256-thread workgroup (8 waves) per batch. Duals in float64 to match the
// reference's numpy float64 u/v. Argmin: in-wave shfl reduction + 8-partial
// combine => only 4 block barriers per augmenting step. First-minimum
// index tie-break matches np.argmin.
// ---------------------------------------------------------------------------
__global__ __launch_bounds__(NTl)
void hm_lsa_kernel(const float* __restrict__ costT,
                   float* __restrict__ out, int B)
{
    __shared__ double s_rv[8];
    __shared__ int    s_ri[8];
    __shared__ double s_v[Qn + 1];
    __shared__ double s_minv[Qn + 1];
    __shared__ double s_u[Nn + 1];
    __shared__ int    s_p[Qn + 1];
    __shared__ int    s_way[Qn + 1];
    __shared__ unsigned char s_used[Qn + 1];
    __shared__ int    s_j0;
    __shared__ double s_delta;
    __shared__ int    s_j1;
    __shared__ int    s_q_of_t[Nn];

    const int tid = threadIdx.x;
    const int b   = blockIdx.x;
    const double DINF = 1.0e300;
    const float* Cb = costT + (size_t)b * Nn * Qn;

    for (int j = tid; j <= Qn; j += NTl) { s_v[j] = 0.0; s_p[j] = 0; }
    if (tid <= Nn) s_u[tid] = 0.0;
    __syncthreads();

    for (int i = 1; i <= Nn; ++i) {
        for (int j = tid; j <= Qn; j += NTl) { s_minv[j] = DINF; s_used[j] = 0; }
        if (tid == 0) { s_p[0] = i; s_j0 = 0; }
        __syncthreads();

        int j1 = 0;
        for (;;) {
            const int j0 = s_j0;
            if (tid == 0) s_used[j0] = 1;
            __syncthreads();                                    // A

            const int i0 = s_p[j0];
            const double ui0 = s_u[i0];
            const float* crow = Cb + (size_t)(i0 - 1) * Qn;

            // scan owned columns (used[0] is always 1 here, so j==0 is inert)
            double best = DINF; int bidx = 0x7fffffff;
            for (int j = tid; j <= Qn; j += NTl) {
                if (!s_used[j]) {
                    const double cur = (double)crow[j - 1] - ui0 - s_v[j];
                    double mv = s_minv[j];
                    if (cur < mv) { mv = cur; s_minv[j] = cur; s_way[j] = j0; }
                    if (mv < best) { best = mv; bidx = j; }     // first-min per thread
                }
            }
            // in-wave argmin (no barrier)
            #pragma unroll
            for (int off = 16; off; off >>= 1) {
                const double ov = __shfl_xor(best, off, 32);
                const int    oi = __shfl_xor(bidx, off, 32);
                if (ov < best || (ov == best && oi < bidx)) { best = ov; bidx = oi; }
            }
            if ((tid & 31) == 0) { s_rv[tid >> 5] = best; s_ri[tid >> 5] = bidx; }
            __syncthreads();                                    // B
            if (tid < 32) {                                     // wave 0 final combine
                double fv = (tid < 8) ? s_rv[tid] : DINF;
                int    fi = (tid < 8) ? s_ri[tid] : 0x7fffffff;
                #pragma unroll
                for (int off = 4; off; off >>= 1) {
                    const double ov = __shfl_xor(fv, off, 32);
                    const int    oi = __shfl_xor(fi, off, 32);
                    if (ov < fv || (ov == fv && oi < fi)) { fv = ov; fi = oi; }
                }
                if (tid == 0) { s_delta = fv; s_j1 = fi; s_j0 = fi; }
            }
            __syncthreads();                                    // C
            const double delta = s_delta;
            j1 = s_j1;

            for (int j = tid; j <= Qn; j += NTl) {
                if (s_used[j]) { s_u[s_p[j]] += delta; s_v[j] -= delta; }
                else           { s_minv[j]   -= delta; }
            }
            const int pj1 = s_p[j1];
            __syncthreads();                                    // D
            if (pj1 == 0) break;                                // uniform
        }
        if (tid == 0) {                                         // augment along `way`
            int j0 = j1;
            do { const int jp = s_way[j0]; s_p[j0] = s_p[jp]; j0 = jp; } while (j0);
        }
        __syncthreads();
    }

    // Extract assignment: q_of_t[target] = query, then stable-sort by query.
    for (int j = tid; j <= Qn; j += NTl) {
        if (j >= 1) { const int pi = s_p[j]; if (pi) s_q_of_t[pi - 1] = j - 1; }
    }
    __syncthreads();
    if (tid < Nn) {
        const int q = s_q_of_t[tid];
        int rank = 0;
        for (int t = 0; t < Nn; ++t) rank += (s_q_of_t[t] < q);  // distinct queries
        out[(size_t)b * Nn + rank]                  = (float)q;   // rows
        out[(size_t)B * Nn + (size_t)b * Nn + rank] = (float)tid; // cols
    }
}

// ---------------------------------------------------------------------------
extern "C" void kernel_launch(void* const* d_in, const int* in_sizes, int n_in,
                              void* d_out, int out_size, void* d_ws, size_t ws_size,
                              hipStream_t stream)
{
    const float* pred_logits = (const float*)d_in[0];   // (B, 600, 92)
    const float* pred_boxes  = (const float*)d_in[1];   // (B, 600, 4)
    const int*   tgt_labels  = (const int*)  d_in[2];   // (B, 80)
    const float* tgt_boxes   = (const float*)d_in[3];   // (B, 80, 4)
    float*       out         = (float*)d_out;           // rows (B,80) ++ cols (B,80)

    const int B = in_sizes[2] / Nn;

    float* probs = (float*)d_ws;                                    // B*QPAD*92 f32
    float* costT = probs + (size_t)B * QPAD * Cc;                   // B*80*600 f32

    // 1) softmax: one wave per row, 8 waves / block
    {
        const int rows = B * QPAD;
        const int blocks = (rows + 7) / 8;
        hm_softmax_kernel<<<blocks, 256, 0, stream>>>(pred_logits, probs, B);
    }
    // 2) WMMA cost build: one wave per (b, q-tile, n-tile), 8 waves / block
    {
        const int waves = B * (QPAD / 16) * (Nn / 16);
        const int blocks = (waves + 7) / 8;
        hm_cost_kernel<<<blocks, 256, 0, stream>>>(probs, pred_boxes, tgt_labels,
                                                   tgt_boxes, costT, B);
    }
    // 3) Hungarian: one block per batch
    hm_lsa_kernel<<<B, NTl, 0, stream>>>(costT, out, B);
}